// MinibatchDiscrimination_54872502174424
// MI455X (gfx1250) — compile-verified
//
#include <hip/hip_runtime.h>
#include <math.h>

typedef __attribute__((ext_vector_type(2))) float v2f;
typedef __attribute__((ext_vector_type(8))) float v8f;

// Problem constants from the reference: N=256, A=1024, B=128, C=32
#define N_ 256
#define A_ 1024
#define B_ 128
#define C_ 32
#define BC_ (B_ * C_)     // 4096
#define OUTW_ (A_ + B_)   // 1152

// ---------------------------------------------------------------------------
// Kernel 1: M = x @ T   (256x1024) * (1024x4096) in f32 via V_WMMA_F32_16X16X4_F32
// Block = 128 threads (4 waves). Each wave computes a 16-row x 64-col strip,
// i.e. 4 WMMA tiles that share one A fragment per K-step (A reuse x4).
// ---------------------------------------------------------------------------
__global__ __launch_bounds__(128)
void mbd_gemm_wmma_f32(const float* __restrict__ x,
                       const float* __restrict__ T,
                       float* __restrict__ M) {
  const int wave = threadIdx.x >> 5;   // 0..3
  const int lane = threadIdx.x & 31;

  const int strips_per_row = BC_ / 64;             // 64 strips of 64 cols
  const int strip = blockIdx.x * 4 + wave;         // 0..1023
  const int trow  = strip / strips_per_row;        // 0..15
  const int tcol  = strip % strips_per_row;        // 0..63
  const int row0  = trow * 16;
  const int col0  = tcol * 64;

  // ISA layout for 16x4 f32 A: lanes 0-15 -> K={0,1}, lanes 16-31 -> K={2,3}
  const int m     = lane & 15;
  const int khalf = (lane >> 4) << 1;              // 0 or 2

  v8f acc0 = {}, acc1 = {}, acc2 = {}, acc3 = {};

#pragma unroll 2
  for (int k0 = 0; k0 < A_; k0 += 4) {
    // A fragment: x[row0+m, k0+khalf .. +1]  (contiguous pair -> b64 load)
    const float* ap = x + (row0 + m) * A_ + (k0 + khalf);
    v2f a;
    a.x = ap[0];
    a.y = ap[1];

    // B fragments: T viewed as (1024 x 4096); row k, cols col0 + {0,16,32,48} + n
    const float* bp0 = T + (size_t)(k0 + khalf) * BC_ + col0 + m;
    const float* bp1 = bp0 + BC_;  // next K row
    v2f b0; b0.x = bp0[0];  b0.y = bp1[0];
    v2f b1; b1.x = bp0[16]; b1.y = bp1[16];
    v2f b2; b2.x = bp0[32]; b2.y = bp1[32];
    v2f b3; b3.x = bp0[48]; b3.y = bp1[48];

    acc0 = __builtin_amdgcn_wmma_f32_16x16x4_f32(false, a, false, b0, (short)0, acc0, false, false);
    acc1 = __builtin_amdgcn_wmma_f32_16x16x4_f32(false, a, false, b1, (short)0, acc1, false, false);
    acc2 = __builtin_amdgcn_wmma_f32_16x16x4_f32(false, a, false, b2, (short)0, acc2, false, false);
    acc3 = __builtin_amdgcn_wmma_f32_16x16x4_f32(false, a, false, b3, (short)0, acc3, false, false);
  }

  // D layout: VGPR r -> row = r + 8*(lane>>4), col = lane&15
  const int drow = row0 + (lane >> 4) * 8;
  const int dcol = col0 + m;
#pragma unroll
  for (int r = 0; r < 8; ++r) {
    float* dst = M + (size_t)(drow + r) * BC_ + dcol;
    dst[0]  = acc0[r];
    dst[16] = acc1[r];
    dst[32] = acc2[r];
    dst[48] = acc3[r];
  }
}

// ---------------------------------------------------------------------------
// Kernel 2: for each feature b (one block per b):
//   o[n,b] = sum_j exp( -sum_c |M[n,b,c] - M[j,b,c]| )
// 256 threads; 256x32 f32 slice staged in LDS (32KB of 320KB/WGP).
// Inner LDS reads are wave-uniform broadcasts (no bank conflicts).
// ---------------------------------------------------------------------------
__global__ __launch_bounds__(256)
void mbd_l1_exp(const float* __restrict__ M, float* __restrict__ out) {
  __shared__ float lds[N_][C_];
  const int b = blockIdx.x;    // 0..127
  const int n = threadIdx.x;   // 0..255

  float mreg[C_];
  const float4* src = (const float4*)(M + (size_t)n * BC_ + b * C_);
  float4* dstl = (float4*)(&lds[n][0]);
#pragma unroll
  for (int q = 0; q < C_ / 4; ++q) {
    float4 v = src[q];
    dstl[q] = v;
    mreg[q * 4 + 0] = v.x;
    mreg[q * 4 + 1] = v.y;
    mreg[q * 4 + 2] = v.z;
    mreg[q * 4 + 3] = v.w;
  }
  __syncthreads();

  float acc = 0.0f;
  for (int j = 0; j < N_; ++j) {
    float s = 0.0f;
#pragma unroll
    for (int c = 0; c < C_; ++c) {
      s += fabsf(mreg[c] - lds[j][c]);
    }
    acc += __expf(-s);   // includes j==n (exp(0)=1), matching the reference
  }

  out[(size_t)n * OUTW_ + A_ + b] = acc;
}

// ---------------------------------------------------------------------------
// Kernel 3: out[n, 0:1024] = x[n, :]   (concat part)
// ---------------------------------------------------------------------------
__global__ __launch_bounds__(256)
void mbd_copy_x(const float* __restrict__ x, float* __restrict__ out) {
  int i = blockIdx.x * blockDim.x + threadIdx.x;  // 0 .. N_*A_-1
  int n = i >> 10;        // / 1024
  int a = i & (A_ - 1);   // % 1024
  out[(size_t)n * OUTW_ + a] = x[i];
}

// ---------------------------------------------------------------------------
extern "C" void kernel_launch(void* const* d_in, const int* in_sizes, int n_in,
                              void* d_out, int out_size, void* d_ws, size_t ws_size,
                              hipStream_t stream) {
  (void)in_sizes; (void)n_in; (void)out_size; (void)ws_size;
  const float* x = (const float*)d_in[0];   // (256, 1024) f32
  const float* T = (const float*)d_in[1];   // (1024, 128, 32) f32
  float* out = (float*)d_out;               // (256, 1152) f32
  float* M   = (float*)d_ws;                // (256, 4096) f32 scratch = 4 MB

  // GEMM: 1024 waves total, 4 waves/block -> 256 blocks of 128 threads
  mbd_gemm_wmma_f32<<<256, 128, 0, stream>>>(x, T, M);

  // Concat copy (independent of GEMM, same stream serializes harmlessly)
  mbd_copy_x<<<(N_ * A_) / 256, 256, 0, stream>>>(x, out);

  // Pairwise L1 + exp reduce: one block per b
  mbd_l1_exp<<<B_, 256, 0, stream>>>(M, out);
}